// SpectralConv1d_15075335209170
// MI455X (gfx1250) — compile-verified
//
#include <hip/hip_runtime.h>

// Problem constants (from reference): B=2, C_IN=8, C_OUT=8, N=8192, MODES=64
#define NLEN   8192
#define MODES  64
#define ROWS   16           // B*C_IN == B*C_OUT
#define W0     (7.6699039394282058e-4f)   // 2*pi / 8192
#define INV_N  (1.220703125e-4f)          // 1 / 8192

typedef __attribute__((ext_vector_type(2))) float v2f;
typedef __attribute__((ext_vector_type(8))) float v8f;

static __device__ __forceinline__ v8f wmma_f32_16x16x4(v2f a, v2f b, v8f c) {
    // D = A(16x4) * B(4x16) + C(16x16), all f32, wave32
    return __builtin_amdgcn_wmma_f32_16x16x4_f32(
        /*neg_a=*/false, a, /*neg_b=*/false, b,
        /*c_mod=*/(short)0, c, /*reuse_a=*/false, /*reuse_b=*/false);
}

// ---------------------------------------------------------------------------
// Stage 1: Xm(16x64) = X(16x8192) @ C(8192x64),  C[n,m] = cos(2*pi*n*m/N)
// One workgroup, 32 waves: wave = (kslice[0..7], modetile[0..3]).
// Each wave accumulates a 16x16 f32 WMMA tile over its K-slice of 1024,
// partials reduced through LDS.
// ---------------------------------------------------------------------------
__global__ void __launch_bounds__(1024)
dht_forward_modes(const float* __restrict__ x, float* __restrict__ Xm) {
    __shared__ float partial[32 * 256];   // 32 waves x (16x16 tile) = 32 KB

    const int lane = threadIdx.x & 31;
    const int wave = threadIdx.x >> 5;
    const int hi   = lane >> 4;           // lane half selects K pair {0,1} vs {2,3}
    const int lo   = lane & 15;
    const int tile = wave & 3;            // which 16 modes
    const int ks   = wave >> 2;           // which K slice
    const int mode = tile * 16 + lo;      // B-matrix column this lane produces

    v8f acc = {};
    const int k0 = ks * 1024;
    for (int k = k0; k < k0 + 1024; k += 4) {
        const int kb = k + 2 * hi;
        v2f a, b;
        // A fragment: row = lo, K = kb, kb+1 (contiguous -> b64 load)
        a.x = x[lo * NLEN + kb];
        a.y = x[lo * NLEN + kb + 1];
        // B fragment: cos(2*pi*(kb*mode mod N)/N), exact integer range reduction
        b.x = __cosf(W0 * (float)((kb * mode) & (NLEN - 1)));
        b.y = __cosf(W0 * (float)(((kb + 1) * mode) & (NLEN - 1)));
        acc = wmma_f32_16x16x4(a, b, acc);
    }

    // D tile: VGPR j -> row (j + 8*hi), col lo
#pragma unroll
    for (int j = 0; j < 8; ++j)
        partial[wave * 256 + (j + 8 * hi) * 16 + lo] = acc[j];
    __syncthreads();

    // Reduce the 8 K-slices: 1024 threads <-> 1024 outputs (16 x 64)
    const int r = threadIdx.x >> 6;       // row 0..15
    const int m = threadIdx.x & 63;       // mode 0..63
    float s = 0.0f;
#pragma unroll
    for (int sl = 0; sl < 8; ++sl)
        s += partial[(sl * 4 + (m >> 4)) * 256 + r * 16 + (m & 15)];
    Xm[r * 64 + m] = s;
}

// ---------------------------------------------------------------------------
// Stage 2: per-mode 8x8 channel mix (tiny; plain VALU).
// Apacked[r][0:64]  = out_re[r][:],  Apacked[r][64:128] = out_im[r][:]
// weights layout: (C_IN, C_OUT, MODES, 2)
// ---------------------------------------------------------------------------
__global__ void __launch_bounds__(1024)
mode_mix(const float* __restrict__ Xm, const float* __restrict__ w,
         float* __restrict__ Apacked) {
    const int t = threadIdx.x;            // exactly 2*8*64 = 1024 threads
    const int m = t & 63;
    const int o = (t >> 6) & 7;
    const int b = t >> 9;
    float ar = 0.0f, ai = 0.0f;
#pragma unroll
    for (int i = 0; i < 8; ++i) {
        const float xv = Xm[(b * 8 + i) * 64 + m];
        const float wre = w[(((i * 8 + o) * 64 + m) << 1) + 0];
        const float wim = w[(((i * 8 + o) * 64 + m) << 1) + 1];
        ar += xv * wre;
        ai += xv * wim;
    }
    const int r = b * 8 + o;
    Apacked[r * 128 + m]      = ar;
    Apacked[r * 128 + 64 + m] = ai;
}

// ---------------------------------------------------------------------------
// Stage 3: Y(16x8192) = Apacked(16x128) @ Binv(128x8192)
//   Binv[m][n]    =  cos(2*pi*m*n/N)/N   for m in [0,64)
//   Binv[64+m][n] = -sin(2*pi*m*n/N)/N
// 128 blocks x 4 waves; each wave owns one 16-column tile (512 tiles total),
// K=128 -> 32 f32 WMMA steps.  A staged in LDS (8 KB).
// ---------------------------------------------------------------------------
__global__ void __launch_bounds__(128)
dht_inverse(const float* __restrict__ Apacked, float* __restrict__ y) {
    __shared__ float As[ROWS * 128];
    for (int i = threadIdx.x; i < ROWS * 128; i += 128)
        As[i] = Apacked[i];
    __syncthreads();

    const int lane = threadIdx.x & 31;
    const int wave = threadIdx.x >> 5;
    const int hi   = lane >> 4;
    const int lo   = lane & 15;
    const int n    = (blockIdx.x * 4 + wave) * 16 + lo;   // output column

    v8f acc = {};
    for (int k = 0; k < 128; k += 4) {
        const int kb = k + 2 * hi;
        v2f a, b;
        a.x = As[lo * 128 + kb];
        a.y = As[lo * 128 + kb + 1];
        // rows [0,64): cos term; rows [64,128): -sin term (branch is k-uniform)
        if (kb < 64) {
            b.x =  INV_N * __cosf(W0 * (float)((kb * n) & (NLEN - 1)));
            b.y =  INV_N * __cosf(W0 * (float)(((kb + 1) * n) & (NLEN - 1)));
        } else {
            b.x = -INV_N * __sinf(W0 * (float)(((kb - 64) * n) & (NLEN - 1)));
            b.y = -INV_N * __sinf(W0 * (float)(((kb - 63) * n) & (NLEN - 1)));
        }
        acc = wmma_f32_16x16x4(a, b, acc);
    }

#pragma unroll
    for (int j = 0; j < 8; ++j)
        y[(j + 8 * hi) * NLEN + n] = acc[j];
}

// ---------------------------------------------------------------------------
extern "C" void kernel_launch(void* const* d_in, const int* in_sizes, int n_in,
                              void* d_out, int out_size, void* d_ws, size_t ws_size,
                              hipStream_t stream) {
    const float* x = (const float*)d_in[0];   // (2, 8, 8192) f32
    const float* w = (const float*)d_in[1];   // (8, 8, 64, 2) f32
    float* out = (float*)d_out;               // (2, 8, 8192) f32

    float* Xm      = (float*)d_ws;            // 16*64  = 1024 floats
    float* Apacked = Xm + 1024;               // 16*128 = 2048 floats

    dht_forward_modes<<<1, 1024, 0, stream>>>(x, Xm);
    mode_mix<<<1, 1024, 0, stream>>>(Xm, w, Apacked);
    dht_inverse<<<128, 128, 0, stream>>>(Apacked, out);
}